// TransitionMatrixGenerator_29798483099672
// MI455X (gfx1250) — compile-verified
//
#include <hip/hip_runtime.h>
#include <stdint.h>

// ---------------------------------------------------------------------------
// Batched 4x4 matrix exponential (scaling & squaring, 16-term Taylor)
//   in : rate_matrix [64, 32768, 4, 4] f32,  time [64] f32
//   out: expm(rate * time)              [64, 32768, 4, 4] f32
// One matrix per thread, register resident. Degree-16 Taylor evaluated via
// Paterson-Stockmeyer (6 matmuls instead of 15), packed fp32 FMA throughout.
// Input staged to LDS via the gfx1250 async-copy path
// (global_load_async_to_lds_b64, GVS addressing / s_wait_asynccnt).
// Output stored with non-temporal hint (write-once stream).
// ---------------------------------------------------------------------------

typedef float v4f __attribute__((ext_vector_type(4)));

#define MAX_SQ       16
#define BLK          256                     // threads per block = matrices per block
#define MAT_BYTES    64
#define LDS_STRIDE   72                      // 18 dwords: bank-conflict-free reads

struct M4 { v4f r[4]; };

// 1/k! for k = 0..16 (folded to fp32 immediates at compile time)
constexpr double INVF[17] = {
    1.0, 1.0, 1.0 / 2.0, 1.0 / 6.0, 1.0 / 24.0, 1.0 / 120.0, 1.0 / 720.0,
    1.0 / 5040.0, 1.0 / 40320.0, 1.0 / 362880.0, 1.0 / 3628800.0,
    1.0 / 39916800.0, 1.0 / 479001600.0, 1.0 / 6227020800.0,
    1.0 / 87178291200.0, 1.0 / 1307674368000.0, 1.0 / 20922789888000.0};

__device__ __forceinline__ v4f splat_fma(float a, v4f b, v4f c) {
  return __builtin_elementwise_fma((v4f)a, b, c);   // -> v_pk_fma_f32 pairs
}

// z = x @ y
__device__ __forceinline__ M4 mm4(const M4& x, const M4& y) {
  M4 z;
#pragma unroll
  for (int i = 0; i < 4; ++i) {
    v4f acc = x.r[i].x * y.r[0];
    acc = splat_fma(x.r[i].y, y.r[1], acc);
    acc = splat_fma(x.r[i].z, y.r[2], acc);
    acc = splat_fma(x.r[i].w, y.r[3], acc);
    z.r[i] = acc;
  }
  return z;
}

// z = x @ y + b   (add folded into the FMA chain init)
__device__ __forceinline__ M4 mm4_add(const M4& x, const M4& y, const M4& b) {
  M4 z;
#pragma unroll
  for (int i = 0; i < 4; ++i) {
    v4f acc = splat_fma(x.r[i].x, y.r[0], b.r[i]);
    acc = splat_fma(x.r[i].y, y.r[1], acc);
    acc = splat_fma(x.r[i].z, y.r[2], acc);
    acc = splat_fma(x.r[i].w, y.r[3], acc);
    z.r[i] = acc;
  }
  return z;
}

// B = c0*I + c1*A + c2*A2 + c3*A3   (no matmuls)
__device__ __forceinline__ M4 poly3(const M4& A, const M4& A2, const M4& A3,
                                    float c0, float c1, float c2, float c3) {
  M4 b;
#pragma unroll
  for (int i = 0; i < 4; ++i) {
    v4f acc = c1 * A.r[i];
    acc = splat_fma(c2, A2.r[i], acc);
    acc = splat_fma(c3, A3.r[i], acc);
    acc[i] += c0;                                      // diagonal
    b.r[i] = acc;
  }
  return b;
}

__global__ __launch_bounds__(BLK) void expm4_kernel(
    const float* __restrict__ rate, const float* __restrict__ timev,
    float* __restrict__ out) {
  __shared__ __align__(16) unsigned char smem[BLK * LDS_STRIDE];

  const int tid  = threadIdx.x;
  const int wave = tid >> 5;
  const int lane = tid & 31;

  // ---- async-stage this block's 16 KB of matrices into LDS ----------------
  // 8B chunks, GVS addressing: uniform 64-bit base stays in SGPRs, one 32-bit
  // VGPR offset per chunk. Wave w issues chunks [256w, 256w+256) -> exactly
  // its own 32 matrices, so no workgroup barrier, only s_wait_asynccnt.
  {
    const uint64_t gbase = (uint64_t)(uintptr_t)(
        (const uint8_t*)rate + (size_t)blockIdx.x * (BLK * MAT_BYTES));
    // chunk g8(i) = wave*256 + i*32 + lane : global off = g8*8 (256B/instr)
    //   LDS off = (g8>>3)*72 + (g8&7)*8
    const uint32_t voff0 = (uint32_t)(wave * 2048 + lane * 8);
    const uint32_t loff0 =
        (uint32_t)(uintptr_t)(&smem[wave * 2304 + (lane >> 3) * 72 +
                                    (lane & 7) * 8]);
#pragma unroll
    for (int i = 0; i < 8; ++i) {
      const uint32_t voff = voff0 + (uint32_t)i * 256u;   // global += 256/iter
      const uint32_t loff = loff0 + (uint32_t)i * 288u;   // lds    += 288/iter
      asm volatile("global_load_async_to_lds_b64 %0, %1, %2"
                   :: "v"(loff), "v"(voff), "s"(gbase)
                   : "memory");
    }
    asm volatile("s_wait_asynccnt 0" ::: "memory");
  }

  // time index is uniform per block: 32768/256 = 128 blocks per batch row
  const float t = timev[blockIdx.x >> 7];

  // ---- load my matrix from LDS (ds_load_2addr_b64, bank-conflict-free) ----
  M4 A;
  {
    const float2* mp = (const float2*)(&smem[tid * LDS_STRIDE]);
#pragma unroll
    for (int i = 0; i < 4; ++i) {
      const float2 lo = mp[2 * i + 0];
      const float2 hi = mp[2 * i + 1];
      v4f row = {lo.x, lo.y, hi.x, hi.y};
      A.r[i] = row * t;                                  // Qt = Q * time
    }
  }

  // ---- infinity norm & scaling --------------------------------------------
  float nrm = 0.0f;
#pragma unroll
  for (int i = 0; i < 4; ++i) {
    const v4f av = __builtin_elementwise_abs(A.r[i]);
    nrm = fmaxf(nrm, (av.x + av.y) + (av.z + av.w));
  }
  // s = clip(ceil(log2(max(nrm,1e-30)/0.5)), 0, 16)
  float sf = ceilf(__log2f(fmaxf(nrm, 1e-30f)) + 1.0f);
  sf = fminf(fmaxf(sf, 0.0f), (float)MAX_SQ);
  const int   si = (int)sf;
  const float sc = __int_as_float((127 - si) << 23);     // exact 2^-si
#pragma unroll
  for (int i = 0; i < 4; ++i) A.r[i] *= sc;

  // ---- degree-16 Taylor via Paterson-Stockmeyer (6 matmuls) ---------------
  //   p(A) = B0 + B1*X + B2*X^2 + B3*X^3 + c16*X^4,  X = A^4
  //   B_i  = INVF[4i]*I + INVF[4i+1]*A + INVF[4i+2]*A^2 + INVF[4i+3]*A^3
  const M4 A2m = mm4(A, A);
  const M4 A3m = mm4(A2m, A);
  const M4 X   = mm4(A2m, A2m);

  M4 P;
  {
    const M4 B3 = poly3(A, A2m, A3m, (float)INVF[12], (float)INVF[13],
                        (float)INVF[14], (float)INVF[15]);
    const float c16 = (float)INVF[16];
#pragma unroll
    for (int i = 0; i < 4; ++i) P.r[i] = splat_fma(c16, X.r[i], B3.r[i]);
  }
  P = mm4_add(P, X, poly3(A, A2m, A3m, (float)INVF[8], (float)INVF[9],
                          (float)INVF[10], (float)INVF[11]));
  P = mm4_add(P, X, poly3(A, A2m, A3m, (float)INVF[4], (float)INVF[5],
                          (float)INVF[6], (float)INVF[7]));
  P = mm4_add(P, X, poly3(A, A2m, A3m, (float)INVF[0], (float)INVF[1],
                          (float)INVF[2], (float)INVF[3]));

  // ---- undo scaling: square si times (early-exit == reference's where) ----
  for (int i = 0; i < MAX_SQ; ++i) {
    if (!__any(si > i)) break;
    if (si > i) P = mm4(P, P);
  }

  // ---- store (4x b128 non-temporal, 64B per thread, write-once stream) ----
  {
    float* op = out + ((size_t)blockIdx.x * BLK + tid) * 16;
    v4f* ov = (v4f*)op;
#pragma unroll
    for (int i = 0; i < 4; ++i) __builtin_nontemporal_store(P.r[i], &ov[i]);
  }
}

extern "C" void kernel_launch(void* const* d_in, const int* in_sizes, int n_in,
                              void* d_out, int out_size, void* d_ws, size_t ws_size,
                              hipStream_t stream) {
  const float* rate  = (const float*)d_in[0];  // [64, 32768, 4, 4] f32
  const float* timev = (const float*)d_in[1];  // [64] f32
  float* out = (float*)d_out;                  // [64, 32768, 4, 4] f32

  const int nmat   = in_sizes[0] / 16;         // 2,097,152
  const int blocks = nmat / BLK;               // 8192
  expm4_kernel<<<blocks, BLK, 0, stream>>>(rate, timev, out);
}